// UpSampler_46420006535684
// MI455X (gfx1250) — compile-verified
//
#include <hip/hip_runtime.h>

#define K_NN     6
#define D_INN    128
#define D_OUTT   128
#define N_COARSE 4000
#define N_FINE   40000

typedef __attribute__((ext_vector_type(2))) float v2f;
typedef __attribute__((ext_vector_type(8))) float v8f;

// ---------------------------------------------------------------------------
// Kernel 1: brute-force KNN (K=6) over squared L2 distance.
// pos_coarse staged in LDS as SoA (48 KB); every lane of a wave reads the same
// coarse point per iteration -> LDS broadcast, conflict-free.
// Stable tie-break (matches jnp.argsort): insert/bubble only on strict '<',
// and j increases monotonically, so earlier indices win ties.
// ---------------------------------------------------------------------------
__global__ __launch_bounds__(256) void knn_kernel(
    const float* __restrict__ pos_fine,
    const float* __restrict__ pos_coarse,
    int* __restrict__ knn_out)
{
    __shared__ float sx[N_COARSE];
    __shared__ float sy[N_COARSE];
    __shared__ float sz[N_COARSE];

    const int tid = threadIdx.x;
    for (int p = tid; p < N_COARSE; p += 256) {
        sx[p] = pos_coarse[3 * p + 0];
        sy[p] = pos_coarse[3 * p + 1];
        sz[p] = pos_coarse[3 * p + 2];
    }
    __syncthreads();

    const int m = blockIdx.x * 256 + tid;
    if (m >= N_FINE) return;

    const float fx = pos_fine[3 * m + 0];
    const float fy = pos_fine[3 * m + 1];
    const float fz = pos_fine[3 * m + 2];

    float bd[K_NN];
    int   bi[K_NN];
#pragma unroll
    for (int t = 0; t < K_NN; ++t) { bd[t] = 3.4e38f; bi[t] = 0; }

#pragma unroll 4
    for (int j = 0; j < N_COARSE; ++j) {
        const float dx = fx - sx[j];
        const float dy = fy - sy[j];
        const float dz = fz - sz[j];
        const float d2 = fmaf(dx, dx, fmaf(dy, dy, dz * dz));
        if (d2 < bd[K_NN - 1]) {
            bd[K_NN - 1] = d2;
            bi[K_NN - 1] = j;
#pragma unroll
            for (int t = K_NN - 1; t > 0; --t) {
                if (bd[t] < bd[t - 1]) {
                    float td = bd[t]; bd[t] = bd[t - 1]; bd[t - 1] = td;
                    int   ti = bi[t]; bi[t] = bi[t - 1]; bi[t - 1] = ti;
                }
            }
        }
    }

#pragma unroll
    for (int t = 0; t < K_NN; ++t) knn_out[m * K_NN + t] = bi[t];
}

// ---------------------------------------------------------------------------
// Kernel 2: fused gather(6-NN) + mean + fp32 GEMM via V_WMMA_F32_16X16X4_F32.
// One block = 16 output rows x all 128 cols. 8 waves; wave w owns N-tile w.
// W staged in LDS (row stride 130 -> conflict-free B reads), interp tile in
// LDS (row stride 132 -> conflict-free A reads). interp never touches HBM.
// EXEC is all-1s everywhere (2500*16 == 40000 exactly), as WMMA requires.
// ---------------------------------------------------------------------------
#define WSTRIDE 130
#define ISTRIDE 132

__global__ __launch_bounds__(256) void interp_gemm_kernel(
    const float* __restrict__ x_coarse,
    const float* __restrict__ Wmat,
    const float* __restrict__ bias,
    const int*  __restrict__ knn,
    float* __restrict__ out)
{
    __shared__ float wlds[D_INN * WSTRIDE];   // 66560 B
    __shared__ float ilds[16 * ISTRIDE];      //  8448 B
    __shared__ int   klds[16 * K_NN];         //   384 B

    const int tid = threadIdx.x;
    const int m0  = blockIdx.x * 16;

    // Stage W (coalesced global reads, padded LDS rows).
    for (int idx = tid; idx < D_INN * D_OUTT; idx += 256) {
        const int r = idx >> 7;
        const int c = idx & 127;
        wlds[r * WSTRIDE + c] = Wmat[idx];
    }
    // Stage the 16 rows' neighbor indices once.
    if (tid < 16 * K_NN) klds[tid] = knn[m0 * K_NN + tid];
    __syncthreads();

    // Gather + mean -> interp tile in LDS (consecutive tids -> consecutive
    // columns -> coalesced x_coarse reads, L2-resident).
    for (int idx = tid; idx < 16 * D_INN; idx += 256) {
        const int r = idx >> 7;
        const int c = idx & 127;
        float acc = 0.0f;
#pragma unroll
        for (int t = 0; t < K_NN; ++t)
            acc += x_coarse[klds[r * K_NN + t] * D_INN + c];
        ilds[r * ISTRIDE + c] = acc * (1.0f / 6.0f);
    }
    __syncthreads();

    const int lane  = tid & 31;          // wave32
    const int wave  = tid >> 5;          // 0..7 -> N-tile
    const int n0    = wave * 16;
    const int l16   = lane & 15;         // M for A, N for B/C/D
    const int khalf = (lane < 16) ? 0 : 2;
    const int mhalf = (lane < 16) ? 0 : 8;

    // C init = bias broadcast down the column (bias depends only on N).
    v8f acc;
    const float bv = bias[n0 + l16];
#pragma unroll
    for (int v = 0; v < 8; ++v) acc[v] = bv;

#pragma unroll 4
    for (int k0 = 0; k0 < D_INN; k0 += 4) {
        v2f afrag, bfrag;
        // A 16x4 f32: lane holds row l16; VGPR v -> K = k0 + khalf + v
        afrag[0] = ilds[l16 * ISTRIDE + k0 + khalf + 0];
        afrag[1] = ilds[l16 * ISTRIDE + k0 + khalf + 1];
        // B 4x16 f32: lane holds col l16; VGPR v -> K = k0 + khalf + v
        bfrag[0] = wlds[(k0 + khalf + 0) * WSTRIDE + n0 + l16];
        bfrag[1] = wlds[(k0 + khalf + 1) * WSTRIDE + n0 + l16];
        acc = __builtin_amdgcn_wmma_f32_16x16x4_f32(
            false, afrag, false, bfrag, (short)0, acc, false, false);
    }

    // D 16x16 f32: VGPR v -> row mhalf+v, col l16.
#pragma unroll
    for (int v = 0; v < 8; ++v)
        out[(m0 + mhalf + v) * D_OUTT + n0 + l16] = acc[v];
}

// ---------------------------------------------------------------------------
extern "C" void kernel_launch(void* const* d_in, const int* in_sizes, int n_in,
                              void* d_out, int out_size, void* d_ws, size_t ws_size,
                              hipStream_t stream)
{
    const float* x_coarse   = (const float*)d_in[0];
    const float* pos_coarse = (const float*)d_in[1];
    const float* pos_fine   = (const float*)d_in[2];
    const float* Wmat       = (const float*)d_in[3];
    const float* bias       = (const float*)d_in[4];
    float* out = (float*)d_out;
    int*   knn = (int*)d_ws;   // N_FINE * K_NN ints = 960 KB scratch

    knn_kernel<<<(N_FINE + 255) / 256, 256, 0, stream>>>(pos_fine, pos_coarse, knn);
    interp_gemm_kernel<<<N_FINE / 16, 256, 0, stream>>>(x_coarse, Wmat, bias, knn, out);
}